// AttentionAggregator_89404039233611
// MI455X (gfx1250) — compile-verified
//
#include <hip/hip_runtime.h>

// ---------------------------------------------------------------------------
// AttentionAggregator for MI455X (gfx1250, wave32).
// N = 100000 (divisible by 16), E = 1600000, DIN = DOUT = 128, H = 4, DH = 32.
//
// Kernel 1: fused dual GEMM (vecs@W0, vecs@W1flat) via V_WMMA_F32_16X16X4_F32,
//           + relu/bias + row-norm of ret_self -> d_out, attention logits and
//           vw -> workspace, zero-init agg.
// Kernel 2: edge scatter with relu attention and global f32 atomics.
// Kernel 3: relu(agg)+b1, row-norm, accumulate into d_out.
// ---------------------------------------------------------------------------

typedef float v2f __attribute__((ext_vector_type(2)));
typedef float v8f __attribute__((ext_vector_type(8)));

#define LDS_STRIDE 132   // 128 + 4 pad -> conflict-free strided fragment reads
#define EPS 1e-9f

__global__ __launch_bounds__(256) void fused_gemm_kernel(
    const float* __restrict__ vecs, const float* __restrict__ W0,
    const float* __restrict__ b0,   const float* __restrict__ W1,
    const float* __restrict__ att0, const float* __restrict__ att1,
    const float* __restrict__ att_b0, const float* __restrict__ att_b1,
    const float* __restrict__ off0, const float* __restrict__ sc0,
    float* __restrict__ vw_g, float* __restrict__ agg_g,
    float* __restrict__ a_self_g, float* __restrict__ a_neigh_g,
    float* __restrict__ out, int N)
{
    __shared__ float atile[16 * LDS_STRIDE];   // vecs tile   (16 x 128)
    __shared__ float stile[16 * LDS_STRIDE];   // relu(x@W0+b0)
    __shared__ float vwtile[16 * LDS_STRIDE];  // x@W1flat

    const int tid  = threadIdx.x;
    const int lane = tid & 31;
    const int wave = tid >> 5;        // 8 waves
    const int r0   = blockIdx.x * 16; // N % 16 == 0

    // ---- stage A tile (16x128 fp32) into LDS, float4 coalesced ----
    #pragma unroll
    for (int i = 0; i < 2; ++i) {
        int idx4 = tid + i * 256;              // 0..511 float4s
        int row  = idx4 >> 5;
        int c4   = idx4 & 31;
        float4 v = *reinterpret_cast<const float4*>(
            vecs + (size_t)(r0 + row) * 128 + c4 * 4);
        *reinterpret_cast<float4*>(&atile[row * LDS_STRIDE + c4 * 4]) = v;
    }
    __syncthreads();

    // ---- per-wave 16x16 output tile, columns [wave*16, wave*16+16) ----
    const int m      = lane & 15;      // A-matrix row (M)
    const int khalf  = lane >> 4;      // K-half select per ISA layout
    const int n      = lane & 15;      // B/C column within tile
    const int col    = wave * 16 + n;  // global column 0..127
    const int h      = col >> 5;       // head for W1 column
    const int k32    = col & 31;       // within-head column
    const float* w1col = W1 + (size_t)h * (128 * 32) + k32;  // stride 32 per d

    v8f c0 = {0.f, 0.f, 0.f, 0.f, 0.f, 0.f, 0.f, 0.f};
    v8f c1 = c0;

    #pragma unroll
    for (int kk = 0; kk < 32; ++kk) {
        const int kb = kk * 4 + 2 * khalf;     // K index of first VGPR element
        v2f a, bw0, bw1;
        a.x   = atile[m * LDS_STRIDE + kb];
        a.y   = atile[m * LDS_STRIDE + kb + 1];
        bw0.x = W0[(size_t)kb * 128 + col];
        bw0.y = W0[(size_t)(kb + 1) * 128 + col];
        bw1.x = w1col[(size_t)kb * 32];
        bw1.y = w1col[(size_t)(kb + 1) * 32];
        c0 = __builtin_amdgcn_wmma_f32_16x16x4_f32(false, a, false, bw0,
                                                   (short)0, c0, false, false);
        c1 = __builtin_amdgcn_wmma_f32_16x16x4_f32(false, a, false, bw1,
                                                   (short)0, c1, false, false);
    }

    // ---- spill fragments to LDS (relu+bias for self path) ----
    const float bias = b0[col];
    #pragma unroll
    for (int j = 0; j < 8; ++j) {
        int row = j + 8 * khalf;               // C/D layout: M=j / M=j+8
        stile[row * LDS_STRIDE + col]  = fmaxf(c0[j] + bias, 0.f);
        vwtile[row * LDS_STRIDE + col] = c1[j];
    }
    __syncthreads();

    // ---- attention logits: a_self/a_neigh = vw . att{0,1} + bias ----
    if (tid < 64) {
        int row = tid >> 2, hh = tid & 3;
        float s0 = 0.f, s1 = 0.f;
        #pragma unroll
        for (int j = 0; j < 32; ++j) {
            float v = vwtile[row * LDS_STRIDE + hh * 32 + j];
            s0 += v * att0[hh * 32 + j];
            s1 += v * att1[hh * 32 + j];
        }
        a_self_g [(size_t)(r0 + row) * 4 + hh] = s0 + att_b0[hh];
        a_neigh_g[(size_t)(r0 + row) * 4 + hh] = s1 + att_b1[hh];
    }

    // ---- row-norm of ret_self (wave handles 2 rows, 32 lanes x 4 elems) ----
    #pragma unroll
    for (int rr = 0; rr < 2; ++rr) {
        int row = wave * 2 + rr;
        int c   = lane * 4;
        float4 x = *reinterpret_cast<float4*>(&stile[row * LDS_STRIDE + c]);
        float s  = x.x + x.y + x.z + x.w;
        float sq = x.x * x.x + x.y * x.y + x.z * x.z + x.w * x.w;
        #pragma unroll
        for (int msk = 16; msk >= 1; msk >>= 1) {
            s  += __shfl_xor(s,  msk, 32);
            sq += __shfl_xor(sq, msk, 32);
        }
        float mean = s * (1.f / 128.f);
        float var  = sq * (1.f / 128.f) - mean * mean;
        float rs   = rsqrtf(var + EPS);
        float4 y;
        y.x = sc0[c + 0] * (x.x - mean) * rs + off0[c + 0];
        y.y = sc0[c + 1] * (x.y - mean) * rs + off0[c + 1];
        y.z = sc0[c + 2] * (x.z - mean) * rs + off0[c + 2];
        y.w = sc0[c + 3] * (x.w - mean) * rs + off0[c + 3];
        *reinterpret_cast<float4*>(out + (size_t)(r0 + row) * 128 + c) = y;
    }

    // ---- write vw to workspace, zero the agg buffer for these rows ----
    #pragma unroll
    for (int i = 0; i < 2; ++i) {
        int idx4 = tid + i * 256;
        int row  = idx4 >> 5;
        int c4   = idx4 & 31;
        float4 v = *reinterpret_cast<float4*>(&vwtile[row * LDS_STRIDE + c4 * 4]);
        *reinterpret_cast<float4*>(vw_g  + (size_t)(r0 + row) * 128 + c4 * 4) = v;
        float4 z = {0.f, 0.f, 0.f, 0.f};
        *reinterpret_cast<float4*>(agg_g + (size_t)(r0 + row) * 128 + c4 * 4) = z;
    }
}

// one wave per 4 edges; lanes cover all 128 output features (4 per lane)
__global__ __launch_bounds__(256) void edge_scatter_kernel(
    const float* __restrict__ vals, const int* __restrict__ rows,
    const int* __restrict__ cols,  const float* __restrict__ vw_g,
    const float* __restrict__ a_self_g, const float* __restrict__ a_neigh_g,
    float* __restrict__ agg_g, int E)
{
    const int lane = threadIdx.x & 31;
    const int wave = threadIdx.x >> 5;
    const int h    = lane >> 3;                 // head = (lane*4)/32
    long long ebase = ((long long)blockIdx.x * 8 + wave) * 4;
    #pragma unroll
    for (int i = 0; i < 4; ++i) {
        long long e = ebase + i;
        if (e >= E) return;
        int r = rows[e], c = cols[e];
        float alpha = a_neigh_g[(size_t)c * 4 + h] + a_self_g[(size_t)r * 4 + h];
        alpha = fmaxf(alpha, 0.f) * vals[e];
        float4 v = *reinterpret_cast<const float4*>(
            vw_g + (size_t)c * 128 + lane * 4);
        float* dst = agg_g + (size_t)r * 128 + lane * 4;
        atomicAdd(dst + 0, alpha * v.x);
        atomicAdd(dst + 1, alpha * v.y);
        atomicAdd(dst + 2, alpha * v.z);
        atomicAdd(dst + 3, alpha * v.w);
    }
}

// one wave per row: relu(agg)+b1, row-norm, accumulate into d_out
__global__ __launch_bounds__(256) void finalize_kernel(
    const float* __restrict__ agg_g, const float* __restrict__ b1,
    const float* __restrict__ off1,  const float* __restrict__ sc1,
    float* __restrict__ out, int N)
{
    const int lane = threadIdx.x & 31;
    const int wave = threadIdx.x >> 5;
    const int row  = blockIdx.x * 8 + wave;
    if (row >= N) return;                        // uniform per wave
    const int c = lane * 4;

    float4 g  = *reinterpret_cast<const float4*>(agg_g + (size_t)row * 128 + c);
    float4 bb = *reinterpret_cast<const float4*>(b1 + c);   // b1 flat == col idx
    float4 x;
    x.x = fmaxf(g.x, 0.f) + bb.x;
    x.y = fmaxf(g.y, 0.f) + bb.y;
    x.z = fmaxf(g.z, 0.f) + bb.z;
    x.w = fmaxf(g.w, 0.f) + bb.w;

    float s  = x.x + x.y + x.z + x.w;
    float sq = x.x * x.x + x.y * x.y + x.z * x.z + x.w * x.w;
    #pragma unroll
    for (int msk = 16; msk >= 1; msk >>= 1) {
        s  += __shfl_xor(s,  msk, 32);
        sq += __shfl_xor(sq, msk, 32);
    }
    float mean = s * (1.f / 128.f);
    float var  = sq * (1.f / 128.f) - mean * mean;
    float rs   = rsqrtf(var + EPS);

    float4 sc = *reinterpret_cast<const float4*>(sc1 + c);
    float4 of = *reinterpret_cast<const float4*>(off1 + c);
    float4 o  = *reinterpret_cast<float4*>(out + (size_t)row * 128 + c);
    o.x += sc.x * (x.x - mean) * rs + of.x;
    o.y += sc.y * (x.y - mean) * rs + of.y;
    o.z += sc.z * (x.z - mean) * rs + of.z;
    o.w += sc.w * (x.w - mean) * rs + of.w;
    *reinterpret_cast<float4*>(out + (size_t)row * 128 + c) = o;
}

extern "C" void kernel_launch(void* const* d_in, const int* in_sizes, int n_in,
                              void* d_out, int out_size, void* d_ws, size_t ws_size,
                              hipStream_t stream) {
    const float* vecs     = (const float*)d_in[0];
    const float* adj_vals = (const float*)d_in[1];
    const float* W0       = (const float*)d_in[2];
    const float* b0       = (const float*)d_in[3];
    const float* W1       = (const float*)d_in[4];
    const float* b1       = (const float*)d_in[5];
    const float* att0     = (const float*)d_in[6];
    const float* att1     = (const float*)d_in[7];
    const float* att_b0   = (const float*)d_in[8];
    const float* att_b1   = (const float*)d_in[9];
    const float* off0     = (const float*)d_in[10];
    const float* sc0      = (const float*)d_in[11];
    const float* off1     = (const float*)d_in[12];
    const float* sc1      = (const float*)d_in[13];
    const int*   adj_rows = (const int*)d_in[14];
    const int*   adj_cols = (const int*)d_in[15];
    float* out = (float*)d_out;

    const int N = in_sizes[0] / 128;
    const int E = in_sizes[1];

    float* ws        = (float*)d_ws;
    float* vw_g      = ws;                          // N*128
    float* agg_g     = vw_g + (size_t)N * 128;      // N*128
    float* a_self_g  = agg_g + (size_t)N * 128;     // N*4
    float* a_neigh_g = a_self_g + (size_t)N * 4;    // N*4

    const int nTiles = N / 16;                      // N % 16 == 0 (100000/16)
    fused_gemm_kernel<<<nTiles, 256, 0, stream>>>(
        vecs, W0, b0, W1, att0, att1, att_b0, att_b1, off0, sc0,
        vw_g, agg_g, a_self_g, a_neigh_g, out, N);

    const int nEdgeBlocks = (E + 31) / 32;          // 8 waves * 4 edges/wave
    edge_scatter_kernel<<<nEdgeBlocks, 256, 0, stream>>>(
        adj_vals, adj_rows, adj_cols, vw_g, a_self_g, a_neigh_g, agg_g, E);

    const int nRowBlocks = (N + 7) / 8;             // 8 rows per block
    finalize_kernel<<<nRowBlocks, 256, 0, stream>>>(
        agg_g, b1, off1, sc1, out, N);
}